// SRSChannelEstimator_33277406610013
// MI455X (gfx1250) — compile-verified
//
#include <hip/hip_runtime.h>
#include <math.h>

#define B_    8
#define UP_   8
#define A_    32
#define L_    3264
#define H_    256
#define BSZ_  12
#define KK_   12
#define WIN_  61
#define D0_   (-30)
#define LDIV4_ 816          // L/4
#define F2L_  6528          // 2*L
#define OUT2_ 576           // 4*BSZ*BSZ
#define WFREQ 1.9249162e-3f // 2*pi/L computed below precisely instead

static __device__ __forceinline__ float wconst() {
  return 6.28318530717958647692f / (float)L_;
}
static __device__ __forceinline__ float2 cmul(float2 a, float2 b) {
  return make_float2(a.x*b.x - a.y*b.y, a.x*b.y + a.y*b.x);
}
static __device__ __forceinline__ float2 cmulc(float2 a, float2 b) { // a * conj(b)
  return make_float2(a.x*b.x + a.y*b.y, a.y*b.x - a.x*b.y);
}

// ---------------- Kernel 1: power at window bins (direct DFT) ----------------
// grid = B*UP*61 blocks, 256 threads. thread t: antenna = t>>3, freq offset = t&7.
__global__ void k_power_win(const float* __restrict__ lsr, const float* __restrict__ lsi,
                            const int* __restrict__ shifts, float* __restrict__ pwin) {
  int id   = blockIdx.x;
  int dIdx = id % WIN_;
  int u    = (id / WIN_) % UP_;
  int b    = id / (WIN_ * UP_);
  int nup  = shifts[u];
  int ipk  = ((KK_ - nup) % KK_) * (L_ / KK_);
  int t    = ipk + (dIdx + D0_);
  t %= L_; if (t < 0) t += L_;

  int tid = threadIdx.x;
  int a   = tid >> 3;
  int fo  = tid & 7;
  const float w = wconst();
  const float* rb = lsr + (size_t)(b*A_ + a) * L_;
  const float* ib = lsi + (size_t)(b*A_ + a) * L_;
  float re = 0.f, im = 0.f;
  for (int f = fo; f < L_; f += 8) {
    int rem = (f * t) % L_;           // exact in int32 (< 2^24)
    float s, c;
    sincosf(w * (float)rem, &s, &c);  // ifft: e^{+i*2pi*f*t/L}
    float xr = rb[f], xi = ib[f];
    re += xr*c - xi*s;
    im += xr*s + xi*c;
  }
  // deterministic reduce of the 8 freq-offset lanes per antenna
  for (int off = 4; off > 0; off >>= 1) {
    re += __shfl_down(re, off, 8);
    im += __shfl_down(im, off, 8);
  }
  __shared__ float smag[A_];
  if (fo == 0) smag[a] = re*re + im*im;
  __syncthreads();
  if (tid == 0) {
    float acc = 0.f;
    for (int k = 0; k < A_; ++k) acc += smag[k];   // fixed order -> deterministic
    pwin[id] = acc;
  }
}

// ---------------- Kernel 2: argmax over window -> toff, m --------------------
__global__ void k_peak(const float* __restrict__ pwin, const int* __restrict__ shifts,
                       int* __restrict__ toff, int* __restrict__ mval) {
  int tid = threadIdx.x;
  if (tid >= B_*UP_) return;
  int u = tid % UP_;
  const float* p = pwin + (size_t)tid * WIN_;
  float best = p[0]; int bi = 0;
  for (int d = 1; d < WIN_; ++d) { float v = p[d]; if (v > best) { best = v; bi = d; } }
  int nup = shifts[u];
  int ipk = ((KK_ - nup) % KK_) * (L_ / KK_);
  int to  = bi + D0_;
  toff[tid] = to;
  mval[tid] = to + ipk;
}

// ---------------- Kernel 3: phasor_m, phasor_T -------------------------------
__global__ void k_phasors(const int* __restrict__ toff, const int* __restrict__ mval,
                          float2* __restrict__ phm, float2* __restrict__ pht) {
  int idx = blockIdx.x * blockDim.x + threadIdx.x;
  if (idx >= B_*UP_*L_) return;
  int n  = idx % L_;
  int bu = idx / L_;
  const float w = wconst();
  float s, c;
  int r1 = (mval[bu] * n) % L_; if (r1 < 0) r1 += L_;
  sincosf(w * (float)r1, &s, &c);
  phm[idx] = make_float2(c, s);
  int r2 = (toff[bu] * n) % L_; if (r2 < 0) r2 += L_;
  sincosf(w * (float)r2, &s, &c);
  pht[idx] = make_float2(c, s);
}

// ---------------- Kernel 4: h_avg (shift + 4-tap mean) -----------------------
__global__ void k_havg(const float* __restrict__ lsr, const float* __restrict__ lsi,
                       const float2* __restrict__ phm, float2* __restrict__ havg) {
  int idx = blockIdx.x * blockDim.x + threadIdx.x;
  if (idx >= B_*UP_*A_*LDIV4_) return;
  int j  = idx % LDIV4_;
  int a  = (idx / LDIV4_) % A_;
  int bu = idx / (LDIV4_ * A_);
  int b  = bu / UP_;
  const float*  rb = lsr + (size_t)(b*A_ + a) * L_;
  const float*  ib = lsi + (size_t)(b*A_ + a) * L_;
  const float2* pm = phm + (size_t)bu * L_;
  float2 acc = make_float2(0.f, 0.f);
  for (int c = 0; c < 4; ++c) {
    int l = 4*j + c;
    float2 t = cmul(make_float2(rb[l], ib[l]), pm[l]);
    acc.x += t.x; acc.y += t.y;
  }
  havg[idx] = make_float2(acc.x * 0.25f, acc.y * 0.25f);
}

// jnp.interp with xp = 1.5 + 4*j
static __device__ __forceinline__ float2 interp_at(const float2* __restrict__ row, int x) {
  if (x < 2)        return row[0];
  if (x >= L_ - 2)  return row[LDIV4_ - 1];
  int j = (x - 2) >> 2;
  float t = ((float)x - 1.5f) * 0.25f - (float)j;
  float2 a = row[j], b = row[j+1];
  return make_float2(a.x + t*(b.x - a.x), a.y + t*(b.y - a.y));
}

// ---------------- Kernel 5: total = sum_u h_interp * conj(phasor_m) ----------
__global__ void k_total(const float2* __restrict__ havg, const float2* __restrict__ phm,
                        float2* __restrict__ total) {
  int idx = blockIdx.x * blockDim.x + threadIdx.x;
  if (idx >= B_*A_*L_) return;
  int l = idx % L_;
  int a = (idx / L_) % A_;
  int b = idx / (L_ * A_);
  float2 acc = make_float2(0.f, 0.f);
  for (int u = 0; u < UP_; ++u) {
    int bu = b*UP_ + u;
    float2 hi = interp_at(havg + (size_t)(bu*A_ + a) * LDIV4_, l);
    float2 t  = cmulc(hi, phm[(size_t)bu*L_ + l]);
    acc.x += t.x; acc.y += t.y;
  }
  total[idx] = acc;
}

// ---------------- Kernel 6: h_res (into d_out, interleaved) + energies -------
__global__ void k_hres(const float* __restrict__ lsr, const float* __restrict__ lsi,
                       const float2* __restrict__ havg, const float2* __restrict__ phm,
                       const float2* __restrict__ total, float2* __restrict__ hres,
                       float* __restrict__ energies) {
  int blk = blockIdx.x;          // bu*A + a
  int a   = blk % A_;
  int bu  = blk / A_;
  int b   = bu / UP_;
  const float2* row = havg + (size_t)blk * LDIV4_;
  const float2* pm  = phm  + (size_t)bu * L_;
  const float*  rb  = lsr  + (size_t)(b*A_ + a) * L_;
  const float*  ib  = lsi  + (size_t)(b*A_ + a) * L_;
  const float2* tb  = total + (size_t)(b*A_ + a) * L_;
  float e = 0.f;
  for (int l = threadIdx.x; l < L_; l += blockDim.x) {
    float2 hi  = interp_at(row, l);
    float2 res = make_float2(rb[l] - tb[l].x, ib[l] - tb[l].y);
    float2 rp  = cmul(res, pm[l]);
    float2 h   = make_float2(hi.x + rp.x, hi.y + rp.y);
    hres[(size_t)blk * L_ + l] = h;
    e += h.x*h.x + h.y*h.y;
  }
  __shared__ float sred[256];
  sred[threadIdx.x] = e;
  __syncthreads();
  for (int s = 128; s > 0; s >>= 1) {
    if (threadIdx.x < s) sred[threadIdx.x] += sred[threadIdx.x + s];
    __syncthreads();
  }
  if (threadIdx.x == 0) energies[blk] = sred[0];
}

// ---------------- Kernel 7: argmax antenna + build feat ----------------------
__global__ void k_feat(const float* __restrict__ energies, const float2* __restrict__ hres,
                       float* __restrict__ feat) {
  int bu = blockIdx.x;
  __shared__ int s_ai;
  if (threadIdx.x == 0) {
    const float* e = energies + (size_t)bu * A_;
    float best = e[0]; int bi = 0;
    for (int a = 1; a < A_; ++a) if (e[a] > best) { best = e[a]; bi = a; }
    s_ai = bi;
  }
  __syncthreads();
  const float2* src = hres + (size_t)(bu*A_ + s_ai) * L_;
  float* frow = feat + (size_t)bu * F2L_;
  for (int l = threadIdx.x; l < L_; l += blockDim.x) {
    float2 h = src[l];
    frow[l]      = h.x;
    frow[L_ + l] = h.y;
  }
}

// ---------------- WMMA GEMM: C(64 x N) = A(64 x K) @ B(K x N) + bias ---------
typedef __attribute__((ext_vector_type(2))) float v2f;
typedef __attribute__((ext_vector_type(8))) float v8f;

__global__ void k_gemm_wmma(const float* __restrict__ Amat, const float* __restrict__ Bmat,
                            const float* __restrict__ bias, float* __restrict__ Cmat,
                            int Kdim, int Ndim, int ntilesN, int relu) {
  int lane  = threadIdx.x & 31;
  int wave  = threadIdx.x >> 5;
  int tile  = blockIdx.x * 8 + wave;      // one wave per 16x16 tile; grid sized exactly
  int tm    = tile / ntilesN;
  int tn    = tile % ntilesN;
  int hl    = lane & 15;
  int khalf = (lane >> 4) << 1;           // lanes 0-15: K 0,1 ; lanes 16-31: K 2,3
  int row   = tm*16 + hl;                 // A-frag: M = lane&15 in both halves
  int col   = tn*16 + hl;                 // B/C frag: N = lane&15
  const float* arow = Amat + (size_t)row * Kdim;
  v8f acc = {};
  for (int k = 0; k < Kdim; k += 4) {
    float2 af = *reinterpret_cast<const float2*>(arow + k + khalf);
    v2f av; av.x = af.x; av.y = af.y;
    v2f bv;
    bv.x = Bmat[(size_t)(k + khalf)     * Ndim + col];
    bv.y = Bmat[(size_t)(k + khalf + 1) * Ndim + col];
    acc = __builtin_amdgcn_wmma_f32_16x16x4_f32(false, av, false, bv,
                                                (short)0, acc, false, false);
  }
  int mbase = tm*16 + ((lane >> 4) << 3); // C/D: VGPR r -> M=r (lanes<16) or r+8
  for (int r = 0; r < 8; ++r) {
    float v = acc[r] + bias[col];
    if (relu) v = fmaxf(v, 0.f);
    Cmat[(size_t)(mbase + r) * Ndim + col] = v;
  }
}

// ---------------- Kernel 10: Wmmse = Cm @ inv(Ar Ar^H + I) -------------------
__global__ void k_wmmse(const float* __restrict__ mlp, float2* __restrict__ wm) {
  int row = blockIdx.x;         // 0..63 (b*UP+u)
  int tid = threadIdx.x;        // 0..143
  int i = tid / BSZ_, j = tid % BSZ_;
  __shared__ float2 Cm[BSZ_][BSZ_];
  __shared__ float2 Ar[BSZ_][BSZ_];
  __shared__ float2 G[BSZ_][2*BSZ_];
  __shared__ float2 sf[BSZ_];
  __shared__ float2 spinv;
  __shared__ int    spiv;
  const float* o = mlp + (size_t)row * OUT2_;
  Cm[i][j] = make_float2(o[i*BSZ_ + j],       o[144 + i*BSZ_ + j]);
  Ar[i][j] = make_float2(o[288 + i*BSZ_ + j], o[432 + i*BSZ_ + j]);
  __syncthreads();
  float2 acc = make_float2((i == j) ? 1.f : 0.f, 0.f);   // + I
  for (int k = 0; k < BSZ_; ++k) {
    float2 t = cmulc(Ar[i][k], Ar[j][k]);                // Ar @ Ar^H
    acc.x += t.x; acc.y += t.y;
  }
  G[i][j]        = acc;
  G[i][BSZ_ + j] = make_float2((i == j) ? 1.f : 0.f, 0.f);
  __syncthreads();
  for (int k = 0; k < BSZ_; ++k) {
    if (tid == 0) {                                      // partial pivot (cabs1)
      int p = k; float bv = fabsf(G[k][k].x) + fabsf(G[k][k].y);
      for (int r = k+1; r < BSZ_; ++r) {
        float v = fabsf(G[r][k].x) + fabsf(G[r][k].y);
        if (v > bv) { bv = v; p = r; }
      }
      spiv = p;
    }
    __syncthreads();
    if (tid < 2*BSZ_ && spiv != k) {
      float2 t = G[k][tid]; G[k][tid] = G[spiv][tid]; G[spiv][tid] = t;
    }
    __syncthreads();
    if (tid == 0) {
      float2 z = G[k][k];
      float den = z.x*z.x + z.y*z.y;
      spinv = make_float2(z.x/den, -z.y/den);
    }
    __syncthreads();
    if (tid < 2*BSZ_) G[k][tid] = cmul(G[k][tid], spinv);
    __syncthreads();
    if (tid < BSZ_) sf[tid] = G[tid][k];
    __syncthreads();
    for (int idx = tid; idx < BSZ_*2*BSZ_; idx += 144) {
      int r = idx / (2*BSZ_), c = idx % (2*BSZ_);
      if (r != k) {
        float2 t = cmul(sf[r], G[k][c]);
        G[r][c].x -= t.x; G[r][c].y -= t.y;
      }
    }
    __syncthreads();
  }
  float2 w = make_float2(0.f, 0.f);
  for (int k = 0; k < BSZ_; ++k) {
    float2 t = cmul(Cm[i][k], G[k][BSZ_ + j]);
    w.x += t.x; w.y += t.y;
  }
  wm[(size_t)row * 144 + tid] = w;
}

// ---------------- Kernel 11: in-place 12x12 block filter * conj(phasor_T) ----
__global__ void k_filter(const float2* __restrict__ wm, const float2* __restrict__ pht,
                         float2* __restrict__ hres) {
  int blk = blockIdx.x;          // bu*A + a
  int bu  = blk / A_;
  __shared__ float2 W[BSZ_][BSZ_];
  if (threadIdx.x < 144)
    (&W[0][0])[threadIdx.x] = wm[(size_t)bu * 144 + threadIdx.x];
  __syncthreads();
  float2* base = hres + (size_t)blk * L_;
  const float2* pt = pht + (size_t)bu * L_;
  for (int jb = threadIdx.x; jb < L_/BSZ_; jb += blockDim.x) {
    float2 h[BSZ_];
    for (int j = 0; j < BSZ_; ++j) h[j] = base[jb*BSZ_ + j];   // read before write
    for (int i = 0; i < BSZ_; ++i) {
      float2 acc = make_float2(0.f, 0.f);
      for (int j = 0; j < BSZ_; ++j) {
        float2 t = cmul(W[i][j], h[j]);
        acc.x += t.x; acc.y += t.y;
      }
      base[jb*BSZ_ + i] = cmulc(acc, pt[jb*BSZ_ + i]);
    }
  }
}

extern "C" void kernel_launch(void* const* d_in, const int* in_sizes, int n_in,
                              void* d_out, int out_size, void* d_ws, size_t ws_size,
                              hipStream_t stream) {
  (void)in_sizes; (void)n_in; (void)out_size; (void)ws_size;
  const float* lsr    = (const float*)d_in[0];
  const float* lsi    = (const float*)d_in[1];
  const int*   shifts = (const int*)  d_in[2];
  const float* W1     = (const float*)d_in[3];
  const float* b1     = (const float*)d_in[4];
  const float* W2     = (const float*)d_in[5];
  const float* b2     = (const float*)d_in[6];

  char* ws = (char*)d_ws;
  size_t off = 0;
  auto take = [&](size_t bytes) -> void* {
    void* p = ws + off;
    off = (off + bytes + 255) & ~(size_t)255;
    return p;
  };
  float*  pwin  = (float*) take((size_t)B_*UP_*WIN_ * sizeof(float));
  int*    toffp = (int*)   take((size_t)B_*UP_ * sizeof(int));
  int*    mvalp = (int*)   take((size_t)B_*UP_ * sizeof(int));
  float2* phm   = (float2*)take((size_t)B_*UP_*L_ * sizeof(float2));
  float2* pht   = (float2*)take((size_t)B_*UP_*L_ * sizeof(float2));
  float2* havg  = (float2*)take((size_t)B_*UP_*A_*LDIV4_ * sizeof(float2));
  float2* totb  = (float2*)take((size_t)B_*A_*L_ * sizeof(float2));
  float*  ener  = (float*) take((size_t)B_*UP_*A_ * sizeof(float));
  float*  feat  = (float*) take((size_t)B_*UP_*F2L_ * sizeof(float));
  float*  h1    = (float*) take((size_t)B_*UP_*H_ * sizeof(float));
  float*  mlp   = (float*) take((size_t)B_*UP_*OUT2_ * sizeof(float));
  float2* wmM   = (float2*)take((size_t)B_*UP_*144 * sizeof(float2));

  float2* hres = (float2*)d_out;   // (B,UP,A,L) complex == final (…,L,2) layout

  k_power_win<<<B_*UP_*WIN_, 256, 0, stream>>>(lsr, lsi, shifts, pwin);
  k_peak<<<1, 64, 0, stream>>>(pwin, shifts, toffp, mvalp);
  k_phasors<<<(B_*UP_*L_ + 255)/256, 256, 0, stream>>>(toffp, mvalp, phm, pht);
  k_havg<<<(B_*UP_*A_*LDIV4_ + 255)/256, 256, 0, stream>>>(lsr, lsi, phm, havg);
  k_total<<<(B_*A_*L_ + 255)/256, 256, 0, stream>>>(havg, phm, totb);
  k_hres<<<B_*UP_*A_, 256, 0, stream>>>(lsr, lsi, havg, phm, totb, hres, ener);
  k_feat<<<B_*UP_, 256, 0, stream>>>(ener, hres, feat);
  // GEMM1: 64x6528 @ 6528x256, 4x16=64 tiles -> 8 blocks of 8 waves
  k_gemm_wmma<<<8, 256, 0, stream>>>(feat, W1, b1, h1, F2L_, H_, 16, 1);
  // GEMM2: 64x256 @ 256x576, 4x36=144 tiles -> 18 blocks of 8 waves
  k_gemm_wmma<<<18, 256, 0, stream>>>(h1, W2, b2, mlp, H_, OUT2_, 36, 0);
  k_wmmse<<<B_*UP_, 144, 0, stream>>>(mlp, wmM);
  k_filter<<<B_*UP_*A_, 256, 0, stream>>>(wmM, pht, hres);
}